// SelfAttention_54589034332544
// MI455X (gfx1250) — compile-verified
//
#include <hip/hip_runtime.h>
#include <math.h>

typedef __attribute__((ext_vector_type(2))) float v2f;
typedef __attribute__((ext_vector_type(8))) float v8f;
typedef __attribute__((ext_vector_type(4))) unsigned int v4u;
typedef __attribute__((ext_vector_type(4))) int v4i;
typedef __attribute__((ext_vector_type(8))) int v8i;

#define DM      4096
#define NHEADS  32
#define HEADDIM 128
#define NBATCH  16
#define PCACHE  2048
#define SEQK    (PCACHE + 1)

#define KC      32          // K-chunk per TDM transfer (floats)
#define LROW    34          // LDS row stride in floats (32 + 2 TDM pad DWORDs)
#define NCHUNK  (DM / KC)   // 128

// ---------------------------------------------------------------------------
// GEMM: O[b][n] = sum_k X[b][k] * W[n][k]   (M=16 batches, N=4096, K=4096)
// Block = 8 waves = 128 features. Per K-chunk of 32, the Tensor Data Mover
// DMAs W tile (128x32 f32) and X tile (16x32 f32) into double-buffered LDS
// (TDM pad: +2 DWORDs per 32-DWORD row -> row stride 34, conflict-free
// ds_load_b64). Waves run V_WMMA_F32_16X16X4_F32 from LDS while the TDM
// fetches the next chunk; sync = s_wait_tensorcnt (wave 0) + barrier.
// ---------------------------------------------------------------------------
__global__ __launch_bounds__(256) void gemm16_wmma_tdm(
    const float* __restrict__ X, const float* __restrict__ W,
    float* __restrict__ O)
{
    __shared__ float lw[2][128 * LROW];   // 2 x 17 KB
    __shared__ float lx[2][16 * LROW];    // 2 x ~2 KB

    const int wave    = threadIdx.x >> 5;
    const int lane    = threadIdx.x & 31;
    const int l16     = lane & 15;
    const int hi      = lane >> 4;            // K sub-pair select
    const int n_block = blockIdx.x * 128;

    // Issue both TDM descriptors for K-chunk c into LDS buffer `buf`.
    auto issue = [&](int c, int buf) {
        const v4i z4 = {0, 0, 0, 0};
#if __clang_major__ >= 23
        const v8i z8 = {0, 0, 0, 0, 0, 0, 0, 0};
#endif
        // group1 flags: data_size=4B(2), pad_enable, pad_interval=4 (32 DW),
        // pad_amount=1 (2 DW)
        const int flags = (int)((2u << 16) | (1u << 20) | (4u << 22) | (1u << 25));

        // ---- W tile: 128 rows x 32 cols, row stride 4096 ----
        {
            uint64_t ga  = (uint64_t)(uintptr_t)(W + (size_t)n_block * DM + c * KC);
            uint32_t lds = (uint32_t)(uintptr_t)&lw[buf][0];
            v4u g0 = { 1u, lds, (uint32_t)ga, (uint32_t)(ga >> 32) | (2u << 30) };
            v8i g1 = { flags,
                       (int)(4096u << 16),        // tensor_dim0[15:0] in [31:16]
                       (int)(4096u << 16),        // td0 hi=0 | tensor_dim1 lo
                       (int)((unsigned)KC << 16), // td1 hi=0 | tile_dim0
                       128,                       // tile_dim1 | tile_dim2=0
                       4096,                      // tensor_dim0_stride[31:0]
                       0, 0 };
#if __clang_major__ >= 23
            __builtin_amdgcn_tensor_load_to_lds(g0, g1, z4, z4, z8, 0);
#else
            __builtin_amdgcn_tensor_load_to_lds(g0, g1, z4, z4, 0);
#endif
        }
        // ---- X tile: 16 rows x 32 cols, row stride 4096 ----
        {
            uint64_t ga  = (uint64_t)(uintptr_t)(X + c * KC);
            uint32_t lds = (uint32_t)(uintptr_t)&lx[buf][0];
            v4u g0 = { 1u, lds, (uint32_t)ga, (uint32_t)(ga >> 32) | (2u << 30) };
            v8i g1 = { flags,
                       (int)(4096u << 16),
                       (int)(4096u << 16),
                       (int)((unsigned)KC << 16),
                       16,
                       4096,
                       0, 0 };
#if __clang_major__ >= 23
            __builtin_amdgcn_tensor_load_to_lds(g0, g1, z4, z4, z8, 0);
#else
            __builtin_amdgcn_tensor_load_to_lds(g0, g1, z4, z4, 0);
#endif
        }
    };

    if (wave == 0) {
        issue(0, 0);
        __builtin_amdgcn_s_wait_tensorcnt(0);
    }
    __syncthreads();

    v8f acc = {0.f, 0.f, 0.f, 0.f, 0.f, 0.f, 0.f, 0.f};
    for (int c = 0; c < NCHUNK; ++c) {
        const int buf = c & 1;
        if (wave == 0 && (c + 1) < NCHUNK) issue(c + 1, buf ^ 1);

        // A lane (l16,hi): W row wave*16+l16, cols 4j+2hi..+1
        // B lane (l16,hi): X row l16 (batch), cols 4j+2hi..+1
        const float* lwr = &lw[buf][(wave * 16 + l16) * LROW + 2 * hi];
        const float* lxr = &lx[buf][l16 * LROW + 2 * hi];
#pragma unroll
        for (int j = 0; j < 8; ++j) {
            v2f a = *(const v2f*)(lwr + 4 * j);
            v2f b = *(const v2f*)(lxr + 4 * j);
            acc = __builtin_amdgcn_wmma_f32_16x16x4_f32(false, a, false, b,
                                                        (short)0, acc, false, false);
        }

        if (wave == 0) __builtin_amdgcn_s_wait_tensorcnt(0);
        __syncthreads();
    }

    // D layout: VGPR r, lanes 0-15 -> feature n0+r, batch lane;
    //           lanes 16-31 -> feature n0+r+8, batch lane-16.
    const int n0 = n_block + wave * 16;
#pragma unroll
    for (int r = 0; r < 8; ++r) {
        const int feat = n0 + r + 8 * hi;
        O[(size_t)l16 * DM + feat] = acc[r];
    }
}

// ---------------------------------------------------------------------------
// RoPE (reference semantics: deinterleave even/odd, rotate, concat halves):
//   out[d]      = x[2d]*fr[d] - x[2d+1]*fi[d]
//   out[64 + d] = x[2d]*fi[d] + x[2d+1]*fr[d]
// ---------------------------------------------------------------------------
__global__ void rope_kernel(const float* __restrict__ q_raw,
                            const float* __restrict__ k_raw,
                            const float* __restrict__ fr,
                            const float* __restrict__ fi,
                            float* __restrict__ q_out,
                            float* __restrict__ k_out)
{
    const int idx = blockIdx.x * 256 + threadIdx.x;   // 16*32*64 = 32768 threads
    const int d  = idx & 63;
    const int bh = idx >> 6;                          // b*32 + h
    const float cr = fr[d], ci = fi[d];
    const size_t base = (size_t)bh * HEADDIM;

    float xr = q_raw[base + 2 * d], xi = q_raw[base + 2 * d + 1];
    q_out[base + d]      = xr * cr - xi * ci;
    q_out[base + 64 + d] = xr * ci + xi * cr;

    xr = k_raw[base + 2 * d]; xi = k_raw[base + 2 * d + 1];
    k_out[base + d]      = xr * cr - xi * ci;
    k_out[base + 64 + d] = xr * ci + xi * cr;
}

// ---------------------------------------------------------------------------
// Attention: one workgroup per (b,h). Streams cache_k/cache_v (1.07 GB).
// Phase 1: one wave = one key row (32 lanes x float4 = 512 B, coalesced),
//          dot via wave32 shfl_xor butterfly; scores -> LDS.
// Phase 2: thread d accumulates sum_p probs[p]*V[p][d] (coalesced over d).
// ---------------------------------------------------------------------------
__global__ __launch_bounds__(256) void attn_kernel(
    const float* __restrict__ q,        // [B*H][128] roped
    const float* __restrict__ knew,     // [B*H][128] roped new key (pos P)
    const float* __restrict__ vnew,     // [B][4096] raw v (feature = h*128+d)
    const float* __restrict__ cache_k,  // [B][P][H][Hd]
    const float* __restrict__ cache_v,  // [B][P][H][Hd]
    float* __restrict__ out)            // [B*H][128]
{
    __shared__ float sc[SEQK];
    __shared__ float redmax[8];
    __shared__ float redsum[8];
    __shared__ float o2[256];

    const int bh   = blockIdx.x;
    const int b    = bh >> 5;
    const int h    = bh & 31;
    const int tid  = threadIdx.x;
    const int wave = tid >> 5;
    const int lane = tid & 31;

    const float4 qv = *(const float4*)(q + (size_t)bh * HEADDIM + lane * 4);
    const float scale = 0.088388347648318447f;            // 1/sqrt(128)
    const size_t kvstride = (size_t)NHEADS * HEADDIM;
    const size_t kbase = ((size_t)b * PCACHE * NHEADS + h) * HEADDIM;

    float lmax = -INFINITY;
    for (int p = wave; p < SEQK; p += 8) {
        const float* krow = (p < PCACHE)
                          ? (cache_k + kbase + (size_t)p * kvstride)
                          : (knew + (size_t)bh * HEADDIM);
        const float4 kv = *(const float4*)(krow + lane * 4);
        float dd = qv.x * kv.x + qv.y * kv.y + qv.z * kv.z + qv.w * kv.w;
#pragma unroll
        for (int off = 16; off > 0; off >>= 1) dd += __shfl_xor(dd, off, 32);
        dd *= scale;
        if (lane == 0) sc[p] = dd;
        lmax = fmaxf(lmax, dd);
    }
    if (lane == 0) redmax[wave] = lmax;
    __syncthreads();

    float bmax = redmax[0];
#pragma unroll
    for (int i = 1; i < 8; ++i) bmax = fmaxf(bmax, redmax[i]);

    float lsum = 0.f;
    for (int p = tid; p < SEQK; p += 256) {
        const float e = __expf(sc[p] - bmax);
        sc[p] = e;
        lsum += e;
    }
#pragma unroll
    for (int off = 16; off > 0; off >>= 1) lsum += __shfl_xor(lsum, off, 32);
    if (lane == 0) redsum[wave] = lsum;
    __syncthreads();                             // also fences sc[] exp writes

    float bsum = 0.f;
#pragma unroll
    for (int i = 0; i < 8; ++i) bsum += redsum[i];
    const float inv = 1.0f / bsum;

    const int d    = tid & 127;
    const int half = tid >> 7;
    const size_t vbase = ((size_t)b * PCACHE * NHEADS + h) * HEADDIM + d;
    float acc = 0.f;
    for (int p = half; p < PCACHE; p += 2)
        acc += sc[p] * cache_v[vbase + (size_t)p * kvstride];
    if (half == 0)
        acc += sc[PCACHE] * vnew[(size_t)bh * HEADDIM + d];
    o2[tid] = acc;
    __syncthreads();
    if (half == 0)
        out[(size_t)bh * HEADDIM + d] = (o2[tid] + o2[tid + 128]) * inv;
}

// ---------------------------------------------------------------------------
// Launcher. Workspace layout (floats):
//   [0)        q_raw   16x4096
//   [65536)    k_raw   16x4096
//   [131072)   v_raw   16x4096
//   [196608)   q_rope  16x4096
//   [262144)   k_rope  16x4096
//   [327680)   attn_o  16x4096
// ---------------------------------------------------------------------------
extern "C" void kernel_launch(void* const* d_in, const int* in_sizes, int n_in,
                              void* d_out, int out_size, void* d_ws, size_t ws_size,
                              hipStream_t stream)
{
    const float* x  = (const float*)d_in[0];
    const float* ck = (const float*)d_in[1];
    const float* cv = (const float*)d_in[2];
    const float* fr = (const float*)d_in[3];
    const float* fi = (const float*)d_in[4];
    const float* wq = (const float*)d_in[5];
    const float* wk = (const float*)d_in[6];
    const float* wv = (const float*)d_in[7];
    const float* wo = (const float*)d_in[8];
    // d_in[9] = start_pos (== PCACHE), compile-time constant here.

    float* ws     = (float*)d_ws;
    float* q_raw  = ws;
    float* k_raw  = ws + 65536;
    float* v_raw  = ws + 131072;
    float* q_rope = ws + 196608;
    float* k_rope = ws + 262144;
    float* attn_o = ws + 327680;

    gemm16_wmma_tdm<<<32, 256, 0, stream>>>(x, wq, q_raw);
    gemm16_wmma_tdm<<<32, 256, 0, stream>>>(x, wk, k_raw);
    gemm16_wmma_tdm<<<32, 256, 0, stream>>>(x, wv, v_raw);
    rope_kernel<<<128, 256, 0, stream>>>(q_raw, k_raw, fr, fi, q_rope, k_rope);
    attn_kernel<<<512, 256, 0, stream>>>(q_rope, k_rope, v_raw, ck, cv, attn_o);
    gemm16_wmma_tdm<<<32, 256, 0, stream>>>(attn_o, wo, (float*)d_out);
}